// CentroidAlignmentLoss_549755813958
// MI455X (gfx1250) — compile-verified
//
#include <hip/hip_runtime.h>

// ---------------------------------------------------------------------------
// CentroidAlignmentLoss for MI455X (gfx1250, wave32)
//
// loss = [ sum_i ||x_i||^2/cnt[l_i]  -  sum_c ||sum_c||^2/cnt_c^2 ] / n_unique
//
// Single streaming pass over E (256 MB ~= 11 us @ 23.3 TB/s); per-class
// segment sums are OneHot(f32) x E(f32) via V_WMMA_F32_16X16X4_F32 (exact
// f32). Deterministic: integer atomics only, fixed-order float reductions.
// Double-buffered 32-row windows keep 16 NT loads in flight per wave;
// 8-wave workgroups (two per row slab, split by column half) raise
// occupancy for latency hiding.
// ---------------------------------------------------------------------------

typedef __attribute__((ext_vector_type(2))) float v2f;
typedef __attribute__((ext_vector_type(8))) float v8f;

#define N_ROWS 262144
#define DIMS   256
#define NCLS   100
#define CT     7            // class tiles of 16 -> 112 padded classes
#define CPAD   (CT * 16)    // 112
#define WCH    8            // K-chunks (4 rows each) per window = 32 rows

// ---------------------------------------------------------------------------
__global__ void cal_zero(unsigned* __restrict__ counts) {
    counts[threadIdx.x] = 0u;   // 128 threads
}

// ---------------------------------------------------------------------------
__global__ __launch_bounds__(256)
void cal_hist(const int* __restrict__ labels, unsigned* __restrict__ counts, int n) {
    __shared__ unsigned h[128];
    if (threadIdx.x < 128) h[threadIdx.x] = 0u;
    __syncthreads();
    for (int i = blockIdx.x * blockDim.x + threadIdx.x; i < n;
         i += gridDim.x * blockDim.x)
        atomicAdd(&h[labels[i]], 1u);
    __syncthreads();
    if (threadIdx.x < 128) {
        unsigned v = h[threadIdx.x];
        if (v) atomicAdd(&counts[threadIdx.x], v);
    }
}

// ---------------------------------------------------------------------------
// Main streaming pass. Block = 256 threads = 8 waves. Two blocks per row
// slab: block owns column half (blockIdx.x & 1), wave w owns D-tile
// 8*half + w. Each chunk: 4 rows. B fragment (4x16 f32, 2 VGPRs/lane):
//   vgpr0 holds K = 2*(lane>=16), vgpr1 holds K+1, column N = lane&15.
// A fragment (16x4 one-hot): M = lane&15, same K split.
__global__ __launch_bounds__(256)
void cal_main(const float* __restrict__ E, const int* __restrict__ labels,
              const unsigned* __restrict__ counts,
              float* __restrict__ psum, float* __restrict__ t1part,
              int rows_per_wg)
{
    __shared__ float lds_inv[128];
    __shared__ float lds_red[8];

    const int tid = threadIdx.x;
    if (tid < 128) {
        unsigned c = counts[tid];
        lds_inv[tid] = c ? 1.0f / (float)c : 0.0f;
    }
    __syncthreads();

    const int slab  = blockIdx.x >> 1;
    const int dhalf = blockIdx.x & 1;
    const int wave  = tid >> 5;               // 0..7
    const int lane  = tid & 31;
    const int n     = lane & 15;
    const int khalf = lane >> 4;              // 0: K=0,1   1: K=2,3
    const int col   = ((dhalf << 3) + wave) * 16 + n;
    const int base0 = slab * rows_per_wg;

    v8f acc[CT] = {};                         // 7 class-tile accumulators
    float t1 = 0.0f;

    const float* rowp = E + (size_t)(base0 + 2 * khalf) * DIMS + col;
    const int*   labp = labels + base0;

    // Prologue: load window 0 (16 independent NT loads, one block).
    float bx[WCH], by[WCH];
#pragma unroll
    for (int u = 0; u < WCH; ++u) {
        bx[u] = __builtin_nontemporal_load(rowp + u * 4 * DIMS);
        by[u] = __builtin_nontemporal_load(rowp + u * 4 * DIMS + DIMS);
    }

#pragma unroll 1
    for (int r = 0; r < rows_per_wg; r += 4 * WCH) {
        // ---- prefetch next window (clamped redundant loads on last iter) ----
        const float* nrowp = rowp + 4 * WCH * DIMS;
        const float* pf    = (r + 4 * WCH < rows_per_wg) ? nrowp : rowp;
        float nbx[WCH], nby[WCH];
#pragma unroll
        for (int u = 0; u < WCH; ++u) {
            nbx[u] = __builtin_nontemporal_load(pf + u * 4 * DIMS);
            nby[u] = __builtin_nontemporal_load(pf + u * 4 * DIMS + DIMS);
        }

        // ---- batched scalar label loads for current window (1 kmcnt wait) ----
        int4 lab[WCH];
#pragma unroll
        for (int u = 0; u < WCH; ++u)
            lab[u] = *reinterpret_cast<const int4*>(labp + 4 * u);

        // ---- process current window ----
#pragma unroll
        for (int u = 0; u < WCH; ++u) {
            const unsigned hit = (1u << (lab[u].x >> 4)) | (1u << (lab[u].y >> 4)) |
                                 (1u << (lab[u].z >> 4)) | (1u << (lab[u].w >> 4));
            v2f b;
            b.x = bx[u];
            b.y = by[u];

            const int labx = khalf ? lab[u].z : lab[u].x;
            const int laby = khalf ? lab[u].w : lab[u].y;

            t1 = fmaf(b.x * b.x, lds_inv[labx], t1);
            t1 = fmaf(b.y * b.y, lds_inv[laby], t1);

#pragma unroll
            for (int t = 0; t < CT; ++t) {
                if (hit & (1u << t)) {        // s_bitcmp + s_cbranch per tile
                    const int cls = (t << 4) + n;
                    v2f a;
                    a.x = (labx == cls) ? 1.0f : 0.0f;
                    a.y = (laby == cls) ? 1.0f : 0.0f;
                    acc[t] = __builtin_amdgcn_wmma_f32_16x16x4_f32(
                        false, a, false, b, (short)0, acc[t], false, false);
                }
            }
        }

        // ---- rotate buffers ----
#pragma unroll
        for (int u = 0; u < WCH; ++u) { bx[u] = nbx[u]; by[u] = nby[u]; }
        rowp = nrowp;
        labp += 4 * WCH;
    }

    // Write per-slab partial sums: psum[slab][class][col] (disjoint cols per
    // column-half block, no atomics). C/D layout: VGPR r -> M = r + 8*khalf.
    float* bp = psum + (size_t)slab * (CPAD * DIMS) + col;
#pragma unroll
    for (int t = 0; t < CT; ++t) {
#pragma unroll
        for (int r8 = 0; r8 < 8; ++r8) {
            const int cls = (t << 4) + r8 + (khalf << 3);
            bp[cls * DIMS] = acc[t][r8];
        }
    }

    // Deterministic t1 reduction: fixed xor tree in wave, fixed order across waves.
#pragma unroll
    for (int m = 16; m >= 1; m >>= 1) t1 += __shfl_xor(t1, m, 32);
    if (lane == 0) lds_red[wave] = t1;
    __syncthreads();
    if (tid == 0) {
        float s = 0.0f;
        for (int w = 0; w < 8; ++w) s += lds_red[w];
        t1part[blockIdx.x] = s;
    }
}

// ---------------------------------------------------------------------------
__global__ __launch_bounds__(256)
void cal_reduce(const float* __restrict__ psum, float* __restrict__ sums, int nwg) {
    const int i = blockIdx.x * 256 + threadIdx.x;
    if (i >= CPAD * DIMS) return;
    float s = 0.0f;
    for (int w = 0; w < nwg; ++w)       // fixed order, coalesced across lanes
        s += psum[(size_t)w * (CPAD * DIMS) + i];
    sums[i] = s;
}

// ---------------------------------------------------------------------------
__global__ __launch_bounds__(128)
void cal_final(const float* __restrict__ sums, const unsigned* __restrict__ counts,
               const float* __restrict__ t1part, int nt1, float* __restrict__ out) {
    __shared__ float cred[128];
    __shared__ float tred[128];
    const int c = threadIdx.x;

    float v = 0.0f;
    if (c < NCLS) {
        const unsigned cnt = counts[c];
        if (cnt > 0) {
            float dot = 0.0f;
            const float* p = sums + c * DIMS;
            for (int d = 0; d < DIMS; ++d) { float x = p[d]; dot = fmaf(x, x, dot); }
            const float ic = 1.0f / (float)cnt;
            v = dot * ic * ic;
        }
    }
    cred[c] = v;

    float tv = 0.0f;
    for (int w = c; w < nt1; w += 128) tv += t1part[w];
    tred[c] = tv;
    __syncthreads();

    if (c == 0) {
        float cent = 0.0f, t1 = 0.0f;
        for (int i = 0; i < 128; ++i) { cent += cred[i]; t1 += tred[i]; }
        int nun = 0;
        for (int i = 0; i < NCLS; ++i) nun += (counts[i] > 0);
        out[0] = (t1 - cent) / (float)(nun > 0 ? nun : 1);
    }
}

// ---------------------------------------------------------------------------
extern "C" void kernel_launch(void* const* d_in, const int* in_sizes, int n_in,
                              void* d_out, int out_size, void* d_ws, size_t ws_size,
                              hipStream_t stream) {
    (void)in_sizes; (void)n_in; (void)out_size;
    const float* E      = (const float*)d_in[0];
    const int*   labels = (const int*)d_in[1];   // jnp.int64 -> int32 (JAX x64 off)
    float*       out    = (float*)d_out;

    // Workspace: psum[nwg][112][256] | sums[112*256] | t1part[2*nwg] | counts[128]
    int nwg = 256;
    auto need = [](int w) -> size_t {
        return (size_t)w * CPAD * DIMS * 4 + (size_t)CPAD * DIMS * 4 +
               (size_t)(2 * w) * 4 + 128 * sizeof(unsigned) + 256;
    };
    while (nwg > 32 && need(nwg) > ws_size) nwg >>= 1;
    const int rows_per_wg = N_ROWS / nwg;   // divisible by 32 for all ladder steps

    float*    psum   = (float*)d_ws;
    float*    sums   = psum + (size_t)nwg * CPAD * DIMS;
    float*    t1p    = sums + CPAD * DIMS;
    unsigned* counts = (unsigned*)(t1p + 2 * nwg);

    cal_zero  <<<1, 128, 0, stream>>>(counts);
    cal_hist  <<<256, 256, 0, stream>>>(labels, counts, N_ROWS);
    cal_main  <<<2 * nwg, 256, 0, stream>>>(E, labels, counts, psum, t1p, rows_per_wg);
    cal_reduce<<<(CPAD * DIMS + 255) / 256, 256, 0, stream>>>(psum, sums, nwg);
    cal_final <<<1, 128, 0, stream>>>(sums, counts, t1p, 2 * nwg, out);
}